// UpSampleFP_8813272891491
// MI455X (gfx1250) — compile-verified
//
#include <hip/hip_runtime.h>
#include <math.h>

// Problem constants (match the reference)
#define BB    4
#define NH    16384
#define NL    4096
#define CLOW  256
#define CSKIP 128
#define CIN   384      // CLOW + CSKIP
#define COUT  256

typedef float v2f __attribute__((ext_vector_type(2)));
typedef float v8f __attribute__((ext_vector_type(8)));

// ---------------------------------------------------------------------------
// Kernel 1: 3-NN search + inverse-distance weights.
// One thread per high-res point; xyz_low for the batch cached in LDS (48 KB).
// Scan uses squared distances (monotonic in d), sqrt only for the 3 winners.
// ---------------------------------------------------------------------------
__global__ void __launch_bounds__(256)
knn3_kernel(const float* __restrict__ xyz_low,
            const float* __restrict__ xyz_high,
            int*   __restrict__ out_idx,
            float* __restrict__ out_w) {
    __shared__ float lx[NL];
    __shared__ float ly[NL];
    __shared__ float lz[NL];

    const int t = threadIdx.x;
    const int blocks_per_batch = NH / 256;
    const int b = blockIdx.x / blocks_per_batch;
    const int n = (blockIdx.x % blocks_per_batch) * 256 + t;

    const float* xl = xyz_low + (size_t)b * NL * 3;
    for (int j = t; j < NL; j += 256) {
        lx[j] = xl[3 * j + 0];
        ly[j] = xl[3 * j + 1];
        lz[j] = xl[3 * j + 2];
    }
    __syncthreads();

    const size_t qbase = ((size_t)b * NH + n) * 3;
    const float qx = xyz_high[qbase + 0];
    const float qy = xyz_high[qbase + 1];
    const float qz = xyz_high[qbase + 2];

    float d0 = 1e30f, d1 = 1e30f, d2 = 1e30f;
    int   i0 = 0,     i1 = 0,     i2 = 0;

    for (int j = 0; j < NL; ++j) {
        const float dx = qx - lx[j];
        const float dy = qy - ly[j];
        const float dz = qz - lz[j];
        const float dd = dx * dx + dy * dy + dz * dz;
        if (dd < d0) {
            d2 = d1; i2 = i1;
            d1 = d0; i1 = i0;
            d0 = dd; i0 = j;
        } else if (dd < d1) {
            d2 = d1; i2 = i1;
            d1 = dd; i1 = j;
        } else if (dd < d2) {
            d2 = dd; i2 = j;
        }
    }

    // Reference: d = sqrt(max(d2,1e-12)); knn_d = max(d,1e-8); w = 1/knn_d; normalize.
    float w0 = 1.0f / fmaxf(sqrtf(fmaxf(d0, 1e-12f)), 1e-8f);
    float w1 = 1.0f / fmaxf(sqrtf(fmaxf(d1, 1e-12f)), 1e-8f);
    float w2 = 1.0f / fmaxf(sqrtf(fmaxf(d2, 1e-12f)), 1e-8f);
    const float inv_s = 1.0f / (w0 + w1 + w2);
    w0 *= inv_s; w1 *= inv_s; w2 *= inv_s;

    const size_t o = ((size_t)b * NH + n) * 3;
    out_idx[o + 0] = i0;  out_idx[o + 1] = i1;  out_idx[o + 2] = i2;
    out_w[o + 0]   = w0;  out_w[o + 1]   = w1;  out_w[o + 2]   = w2;
}

// ---------------------------------------------------------------------------
// Kernel 2: fused interp-gather + concat + Linear via V_WMMA_F32_16X16X4_F32.
// 256 threads (8 wave32) per WG; WG computes a 64-row x 256-col output tile.
// K=384 processed in chunks of 32 staged in LDS (A: interp/skip on the fly,
// B: coalesced W slice). Wave w: row-block (w&3), 8 col tiles (w>>2)*8 .. +7.
// ---------------------------------------------------------------------------
__global__ void __launch_bounds__(256)
fp_fused_gemm_kernel(const float* __restrict__ feat_low,    // (B, CLOW, NL)
                     const float* __restrict__ feat_skip,   // (B, NH, CSKIP)
                     const float* __restrict__ Wmat,        // (COUT, CIN)
                     const float* __restrict__ bias,        // (COUT)
                     const int*   __restrict__ knn_idx,     // (B*NH, 3)
                     const float* __restrict__ knn_w,       // (B*NH, 3)
                     float*       __restrict__ out) {       // (B*NH, COUT)
    __shared__ float Asm[64 * 33];    // 64 rows x 32 K, pad stride 33
    __shared__ float Bsm[32 * 257];   // 32 K x 256 N (W^T slice), pad stride 257

    const int t     = threadIdx.x;
    const int grow0 = blockIdx.x * 64;           // global row base (b*NH + n)
    const int b     = grow0 / NH;

    // --- per-thread staging identity for A: row r = t&63, K-lane group q ---
    const int r    = t & 63;
    const int q    = t >> 6;                     // 0..3
    const int grow = grow0 + r;

    const int   gi0 = knn_idx[(size_t)grow * 3 + 0];
    const int   gi1 = knn_idx[(size_t)grow * 3 + 1];
    const int   gi2 = knn_idx[(size_t)grow * 3 + 2];
    const float gw0 = knn_w[(size_t)grow * 3 + 0];
    const float gw1 = knn_w[(size_t)grow * 3 + 1];
    const float gw2 = knn_w[(size_t)grow * 3 + 2];

    const float* FL = feat_low  + (size_t)b * CLOW * NL;
    const float* SK = feat_skip + (size_t)grow * CSKIP;

    // --- WMMA lane identity ---
    const int wv   = t >> 5;                     // wave 0..7
    const int lane = t & 31;
    const int lo   = lane & 15;
    const int hi   = lane >> 4;
    const int rb   = wv & 3;                     // row block
    const int cbb  = (wv >> 2) * 8;              // first of 8 col blocks

    v8f acc[8] = {};

    for (int kk = 0; kk < CIN; kk += 32) {
        __syncthreads();   // previous chunk's consumers done before overwrite

        // Stage A chunk: interp features (gather) or skip features.
        if (kk < CLOW) {
#pragma unroll
            for (int i = 0; i < 8; ++i) {
                const int kloc = q + 4 * i;
                const int c    = kk + kloc;
                Asm[r * 33 + kloc] = gw0 * FL[(size_t)c * NL + gi0]
                                   + gw1 * FL[(size_t)c * NL + gi1]
                                   + gw2 * FL[(size_t)c * NL + gi2];
            }
        } else {
#pragma unroll
            for (int i = 0; i < 8; ++i) {
                const int kloc = q + 4 * i;
                Asm[r * 33 + kloc] = SK[(kk - CLOW) + kloc];
            }
        }

        // Stage B chunk (W^T slice): lanes read 32 consecutive floats of a W row.
        {
            const int kloc = lane;               // 0..31 within wave
            const int nb   = wv;                 // 0..7
#pragma unroll
            for (int j = 0; j < 32; ++j) {
                const int n = nb + 8 * j;        // covers 0..255
                Bsm[kloc * 257 + n] = Wmat[(size_t)n * CIN + (kk + kloc)];
            }
        }
        __syncthreads();

        // 8 K-steps of 4, 8 col tiles each -> 64 WMMAs per chunk per wave.
#pragma unroll
        for (int k0 = 0; k0 < 32; k0 += 4) {
            v2f a;
            a.x = Asm[(rb * 16 + lo) * 33 + k0 + 2 * hi];
            a.y = Asm[(rb * 16 + lo) * 33 + k0 + 2 * hi + 1];
#pragma unroll
            for (int j = 0; j < 8; ++j) {
                const int n = (cbb + j) * 16 + lo;
                v2f bf;
                bf.x = Bsm[(k0 + 2 * hi) * 257 + n];
                bf.y = Bsm[(k0 + 2 * hi + 1) * 257 + n];
                acc[j] = __builtin_amdgcn_wmma_f32_16x16x4_f32(
                    /*neg_a=*/false, a, /*neg_b=*/false, bf,
                    /*c_mod=*/(short)0, acc[j],
                    /*reuse_a=*/false, /*reuse_b=*/false);
            }
        }
    }

    // Epilogue: bias add + store. D layout: acc[v] holds row (v + 8*hi), col lo.
#pragma unroll
    for (int j = 0; j < 8; ++j) {
        const int n  = (cbb + j) * 16 + lo;
        const float bv = bias[n];
#pragma unroll
        for (int v = 0; v < 8; ++v) {
            const int row = grow0 + rb * 16 + v + 8 * hi;
            out[(size_t)row * COUT + n] = acc[j][v] + bv;
        }
    }
}

// ---------------------------------------------------------------------------
// Host launcher
// ---------------------------------------------------------------------------
extern "C" void kernel_launch(void* const* d_in, const int* in_sizes, int n_in,
                              void* d_out, int out_size, void* d_ws, size_t ws_size,
                              hipStream_t stream) {
    const float* xyz_low   = (const float*)d_in[0];   // (B, NL, 3)
    const float* xyz_high  = (const float*)d_in[1];   // (B, NH, 3)
    const float* feat_low  = (const float*)d_in[2];   // (B, CLOW, NL)
    const float* feat_skip = (const float*)d_in[3];   // (B, NH, CSKIP)
    const float* Wmat      = (const float*)d_in[4];   // (COUT, CIN)
    const float* bias      = (const float*)d_in[5];   // (COUT)
    float*       out       = (float*)d_out;           // (B, NH, COUT)

    int*   knn_idx = (int*)d_ws;
    float* knn_w   = (float*)((char*)d_ws + (size_t)BB * NH * 3 * sizeof(int));

    // Kernel 1: one thread per high point.
    dim3 grid1(BB * (NH / 256));
    knn3_kernel<<<grid1, 256, 0, stream>>>(xyz_low, xyz_high, knn_idx, knn_w);

    // Kernel 2: 64 output rows per workgroup over all B*NH rows.
    dim3 grid2((BB * NH) / 64);
    fp_fused_gemm_kernel<<<grid2, 256, 0, stream>>>(feat_low, feat_skip, Wmat,
                                                    bias, knn_idx, knn_w, out);
}